// QuantumRegression__gen334_65481071407560
// MI455X (gfx1250) — compile-verified
//
#include <hip/hip_runtime.h>
#include <hip/hip_bf16.h>

// ============================================================================
// 16-qubit batched statevector simulator for MI455X (gfx1250).
//
// Strategy: factor the 16 wires into 4 nibbles of 4 qubits. Per layer:
//   - fused 16x16 complex unitary per nibble  -> f32 WMMA (16x16x4) GEMM pass
//   - intra-nibble CNOT chain                 -> prefix-XOR output scatter
//   - cross-nibble CNOT                       -> conditional XOR 0xF on index
// psi kept as interleaved complex float2, ping-ponged in workspace
// (2 x 128 MB, L2-resident on the 192 MB L2). 16 GEMM passes total,
// ~34 GFLOP f32 on the matrix units, ~4 GB of L2 traffic.
// Workspace requirement: ~268.7 MB.
// ============================================================================

typedef __attribute__((ext_vector_type(2))) float v2f;
typedef __attribute__((ext_vector_type(8))) float v8f;

static __device__ __forceinline__ v8f wmma4(v2f a, v2f b, v8f c) {
  // D = A(16x4 f32) x B(4x16 f32) + C(16x16 f32)
  return __builtin_amdgcn_wmma_f32_16x16x4_f32(
      /*neg_a=*/false, a, /*neg_b=*/false, b,
      /*c_mod=*/(short)0, c, /*reuse_a=*/false, /*reuse_b=*/false);
}

static __device__ __forceinline__ float2 cmul(float2 a, float2 b) {
  return make_float2(a.x * b.x - a.y * b.y, a.x * b.y + a.y * b.x);
}

// ---------------------------------------------------------------------------
// Build fused 16x16 complex matrices: S[layer][group] = U_{4g} (x) ... (x) U_{4g+3}
// with U = RZ(rz) RY(ry) RX(rx). Qubit (4g+k) maps to nibble bit (3-k).
// 4096 threads: one matrix entry each.
// ---------------------------------------------------------------------------
__global__ void k_build_S(const float* __restrict__ params, float2* __restrict__ Smat) {
  int idx = blockIdx.x * blockDim.x + threadIdx.x;  // [0, 4096)
  int layer = idx >> 10;
  int g = (idx >> 8) & 3;
  int i = (idx >> 4) & 15;
  int ip = idx & 15;
  float2 prod = make_float2(1.f, 0.f);
  for (int k = 0; k < 4; ++k) {
    const float* th = params + (layer * 16 + (g * 4 + k)) * 3;
    float a = th[0] * 0.5f, b = th[1] * 0.5f, c = th[2] * 0.5f;
    float cx = cosf(a), sx = sinf(a);
    float cy = cosf(b), sy = sinf(b);
    float cc = cosf(c), sc = sinf(c);
    int bi = (i >> (3 - k)) & 1;
    int bip = (ip >> (3 - k)) & 1;
    // M1 = RY @ RX
    float2 m1;
    if (bi == 0) {
      m1 = (bip == 0) ? make_float2(cy * cx, sy * sx)
                      : make_float2(-sy * cx, -cy * sx);
    } else {
      m1 = (bip == 0) ? make_float2(sy * cx, -cy * sx)
                      : make_float2(cy * cx, -sy * sx);
    }
    // RZ phase: row0 *= e^{-ic}, row1 *= e^{+ic}
    float2 ph = make_float2(cc, bi ? sc : -sc);
    prod = cmul(prod, cmul(ph, m1));
  }
  Smat[idx] = prod;  // layout: [(layer*4+g)*256 + i*16 + ip]
}

// ---------------------------------------------------------------------------
// psi0 = real input, imag 0 (interleaved complex float2)
// ---------------------------------------------------------------------------
__global__ void k_init(const float* __restrict__ s, float2* __restrict__ psi) {
  int i = blockIdx.x * blockDim.x + threadIdx.x;  // 16,777,216 total
  psi[i] = make_float2(s[i], 0.f);
}

// ---------------------------------------------------------------------------
// Sign table: s(i) = sum_w w_head[w] * (1 - 2*bit_{15-w}(i))
// ---------------------------------------------------------------------------
__global__ void k_sign(const float* __restrict__ wh, float* __restrict__ stab) {
  int i = blockIdx.x * blockDim.x + threadIdx.x;  // [0, 65536)
  float s = 0.f;
#pragma unroll
  for (int w = 0; w < 16; ++w) {
    s += wh[w] * (((i >> (15 - w)) & 1) ? -1.f : 1.f);
  }
  stab[i] = s;
}

// ---------------------------------------------------------------------------
// Axis pass, nibble stride st = 1<<stsh (4096 / 256 / 16):
//   out[base + (p4(m)^mask)*st + n] = sum_k S[m,k] * in[base + k*st + n]
// One 16x16 complex output tile per wave; 4 f32 WMMAs per K-step, K=16.
// mask = 0xF when the controlling (already-permuted) bystander bit is set.
// ---------------------------------------------------------------------------
__global__ void __launch_bounds__(256) k_pass_hi(const float2* __restrict__ in,
                                                 float2* __restrict__ out,
                                                 const float2* __restrict__ S,
                                                 int stsh, int ctrlEn) {
  __shared__ float2 sS[256];
  sS[threadIdx.x] = S[threadIdx.x];
  __syncthreads();

  const int lane = threadIdx.x & 31;
  const int t = blockIdx.x * 8 + (threadIdx.x >> 5);  // tile id [0, 65536)
  const int b = t >> 8;
  const int nb = t & 255;
  const int st = 1 << stsh;
  const int hi = nb >> (stsh - 4);
  const int lo = (nb & ((1 << (stsh - 4)) - 1)) << 4;
  const int base = (b << 16) + (hi << (stsh + 4)) + lo;
  const int mask = (hi & ctrlEn) ? 15 : 0;  // cross-CNOT fold (target nibble)
  const int n = lane & 15;                  // A row (M) and B column (N)
  const int hs = (lane >> 4) << 1;          // K sub-offset per lane half

  v8f dre = {0.f, 0.f, 0.f, 0.f, 0.f, 0.f, 0.f, 0.f};
  v8f dim = {0.f, 0.f, 0.f, 0.f, 0.f, 0.f, 0.f, 0.f};

#pragma unroll
  for (int k0 = 0; k0 < 4; ++k0) {
    const int kb = k0 * 4 + hs;
    float2 s0 = sS[n * 16 + kb];       // A = S (M=new nibble, K=old nibble)
    float2 s1 = sS[n * 16 + kb + 1];
    v2f are = {s0.x, s1.x};
    v2f aim = {s0.y, s1.y};
    v2f aimn = -aim;
    float2 p0 = in[base + kb * st + n];        // B = psi tile (K x N)
    float2 p1 = in[base + (kb + 1) * st + n];
    v2f bre = {p0.x, p1.x};
    v2f bim = {p0.y, p1.y};
    dre = wmma4(aimn, bim, dre);  // Dre += Ar*Br - Ai*Bi
    dre = wmma4(are, bre, dre);
    dim = wmma4(are, bim, dim);   // Dim += Ar*Bi + Ai*Br
    dim = wmma4(aim, bre, dim);
  }

  const int halfAdd = (lane >> 4) << 3;
#pragma unroll
  for (int v = 0; v < 8; ++v) {
    int m2 = v + halfAdd;              // D row = new nibble index
    int mo = m2 ^ (m2 >> 1);           // intra-nibble CNOT chain: prefix-XOR
    mo ^= (mo >> 2);
    int iout = mo ^ mask;
    out[base + iout * st + n] = make_float2(dre[v], dim[v]);
  }
}

// ---------------------------------------------------------------------------
// Last-nibble pass (stride 1): D(16 j-rows x 16 new-nibble) = Psi_tile x S^T.
// Cross-CNOT control = LSB of the j index (nibble n1, post-permutation).
// ---------------------------------------------------------------------------
__global__ void __launch_bounds__(256) k_pass_n0(const float2* __restrict__ in,
                                                 float2* __restrict__ out,
                                                 const float2* __restrict__ S) {
  __shared__ float2 sS[256];
  sS[threadIdx.x] = S[threadIdx.x];
  __syncthreads();

  const int lane = threadIdx.x & 31;
  const int t = blockIdx.x * 8 + (threadIdx.x >> 5);  // [0, 65536)
  const int b = t >> 8;
  const int jb = (t & 255) << 4;      // 16 j-rows per tile
  const int base = (b << 16) + (jb << 4);
  const int n = lane & 15;            // A row (j) and B column (new nibble)
  const int hs = (lane >> 4) << 1;

  v8f dre = {0.f, 0.f, 0.f, 0.f, 0.f, 0.f, 0.f, 0.f};
  v8f dim = {0.f, 0.f, 0.f, 0.f, 0.f, 0.f, 0.f, 0.f};

  const float4* in4 = reinterpret_cast<const float4*>(in);
#pragma unroll
  for (int k0 = 0; k0 < 4; ++k0) {
    const int kb = k0 * 4 + hs;
    // A = psi tile: two adjacent complex amplitudes -> one b128 load
    float4 aq = in4[(base + n * 16 + kb) >> 1];
    v2f are = {aq.x, aq.z};
    v2f aim = {aq.y, aq.w};
    v2f aimn = -aim;
    // B = S^T: T[k, n] = S[n, k]
    float2 s0 = sS[n * 16 + kb];
    float2 s1 = sS[n * 16 + kb + 1];
    v2f bre = {s0.x, s1.x};
    v2f bim = {s0.y, s1.y};
    dre = wmma4(aimn, bim, dre);
    dre = wmma4(are, bre, dre);
    dim = wmma4(are, bim, dim);
    dim = wmma4(aim, bre, dim);
  }

  int mo = n ^ (n >> 1);  // p4 of the new-nibble (N) index
  mo ^= (mo >> 2);
  const int halfAdd = (lane >> 4) << 3;
#pragma unroll
  for (int v = 0; v < 8; ++v) {
    int m2 = v + halfAdd;                       // j row within tile
    int iout = mo ^ ((m2 & 1) ? 15 : 0);        // ctrl = LSB of j (nibble n1)
    out[base + m2 * 16 + iout] = make_float2(dre[v], dim[v]);
  }
}

// ---------------------------------------------------------------------------
// out[b] = sum_i |psi[b,i]|^2 * s(i) + b_head
// ---------------------------------------------------------------------------
__global__ void __launch_bounds__(256) k_head(const float2* __restrict__ psi,
                                              const float* __restrict__ stab,
                                              const float* __restrict__ bh,
                                              float* __restrict__ out) {
  __shared__ float red[256];
  const int b = blockIdx.x;
  const int tt = threadIdx.x;
  const float2* p = psi + ((size_t)b << 16);
  float acc = 0.f;
  for (int j = 0; j < 256; ++j) {
    int i = (j << 8) + tt;
    float2 a = p[i];
    acc += (a.x * a.x + a.y * a.y) * stab[i];
  }
  red[tt] = acc;
  __syncthreads();
  for (int s = 128; s > 0; s >>= 1) {
    if (tt < s) red[tt] += red[tt + s];
    __syncthreads();
  }
  if (tt == 0) out[b] = red[0] + bh[0];
}

// ---------------------------------------------------------------------------
extern "C" void kernel_launch(void* const* d_in, const int* in_sizes, int n_in,
                              void* d_out, int out_size, void* d_ws, size_t ws_size,
                              hipStream_t stream) {
  const float* state = (const float*)d_in[0];   // [256, 65536]
  const float* params = (const float*)d_in[1];  // [4, 16, 3]
  const float* w_head = (const float*)d_in[2];  // [16]
  const float* b_head = (const float*)d_in[3];  // [1]
  float* out = (float*)d_out;                   // [256]

  char* ws = (char*)d_ws;
  const size_t BUF_BYTES = (size_t)256 * 65536 * sizeof(float2);  // 128 MB
  float2* buf0 = (float2*)ws;
  float2* buf1 = (float2*)(ws + BUF_BYTES);
  float2* Smat = (float2*)(ws + 2 * BUF_BYTES);                   // 32 KB (4096 float2)
  float* stab = (float*)(ws + 2 * BUF_BYTES + 16384);             // 256 KB
  float2* bufs[2] = {buf0, buf1};

  // Fused nibble unitaries + initial state + sign table
  k_build_S<<<16, 256, 0, stream>>>(params, Smat);
  k_init<<<65536, 256, 0, stream>>>(state, buf0);
  k_sign<<<256, 256, 0, stream>>>(w_head, stab);

  // 4 layers x 4 nibble passes (G0..G3), ping-pong through L2-resident buffers
  int pp = 0;
  for (int layer = 0; layer < 4; ++layer) {
    for (int g = 0; g < 4; ++g) {
      const float2* S = Smat + (layer * 4 + g) * 256;
      const float2* src = bufs[pp & 1];
      float2* dst = bufs[(pp & 1) ^ 1];
      if (g < 3) {
        // g=0: axis n3 (stride 4096, no cross ctrl); g=1: n2 (256); g=2: n1 (16)
        k_pass_hi<<<8192, 256, 0, stream>>>(src, dst, S, 12 - 4 * g, g == 0 ? 0 : 1);
      } else {
        k_pass_n0<<<8192, 256, 0, stream>>>(src, dst, S);
      }
      ++pp;
    }
  }

  // After 16 passes the state is back in buf0
  k_head<<<256, 256, 0, stream>>>(bufs[pp & 1], stab, b_head, out);
}